// VQVAE_84353157693461
// MI455X (gfx1250) — compile-verified
//
#include <hip/hip_runtime.h>
#include <hip/hip_bf16.h>
#include <stdint.h>

// ---------------- problem constants ----------------
#define NUM_ENTRIES   8192
#define CODE_DIM      256
#define N_ROWS        32768          // 32 * 1024
#define N_ELEMS       8388608        // N_ROWS * CODE_DIM
#define TILE_CODES    16
#define NUM_TILES     (NUM_ENTRIES / TILE_CODES)   // 512
#define LDS_STRIDE    260            // 260 mod 64 == 4 -> conflict-free B reads
#define ROWS_PER_BLOCK 128           // 8 waves * 16 rows

typedef __attribute__((ext_vector_type(2))) float v2f;
typedef __attribute__((ext_vector_type(8))) float v8f;
typedef __attribute__((ext_vector_type(4))) int   v4i;

// exact pointer types for the async global->LDS builtin:
//   (v4i addrspace(1)*, v4i addrspace(3)*, imm offset, imm cpol)
typedef __attribute__((address_space(1))) v4i* gas_v4i_ptr;
typedef __attribute__((address_space(3))) v4i* las_v4i_ptr;

// ------------- gfx1250 async global->LDS path (guarded) -------------
#define VQ_ASYNC 0
#if defined(__has_builtin)
#if __has_builtin(__builtin_amdgcn_global_load_async_to_lds_b128) && \
    __has_builtin(__builtin_amdgcn_s_wait_asynccnt)
#undef VQ_ASYNC
#define VQ_ASYNC 1
#endif
#endif

// =====================================================================
// Kernel 1: codebook row squared norms  ||e_k||^2
// one wave per codebook row; 8 f32 per lane, shuffle-reduce
// =====================================================================
__global__ __launch_bounds__(256) void vq_prep(const float* __restrict__ cb,
                                               float* __restrict__ enorm) {
  const int lane = threadIdx.x & 31;
  const int wave = threadIdx.x >> 5;
  const int code = blockIdx.x * 8 + wave;  // grid = 1024 blocks
  const float* row = cb + (size_t)code * CODE_DIM;
  float s = 0.f;
#pragma unroll
  for (int j = 0; j < 8; ++j) {
    float v = row[lane + 32 * j];
    s += v * v;
  }
#pragma unroll
  for (int off = 16; off >= 1; off >>= 1) s += __shfl_xor(s, off, 32);
  if (lane == 0) enorm[code] = s;
}

// =====================================================================
// Kernel 2: distance argmin via fp32 WMMA
//   grid 256 blocks x 256 threads (8 waves); wave owns 16 z rows.
//   A (z rows) pinned in 128 VGPRs; B (codebook tile) double-buffered in LDS
//   via gfx1250 async global->LDS; B LDS reads software-pipelined depth-2.
// =====================================================================
__global__ __launch_bounds__(256) void vq_main(const float* __restrict__ z,
                                               const float* __restrict__ cb,
                                               const float* __restrict__ enorm,
                                               int* __restrict__ out_idx) {
  __shared__ float lds_b[2][TILE_CODES * LDS_STRIDE];

  const int tid     = threadIdx.x;
  const int lane    = tid & 31;
  const int wave    = tid >> 5;
  const int lane_lo = lane & 15;
  const int half    = lane >> 4;
  const int wrow    = blockIdx.x * ROWS_PER_BLOCK + wave * 16;
  const int arow    = wrow + lane_lo;

  // --- preload this wave's 16 z rows: 64 K-chunks of float2 per lane ---
  // A 16x4 f32 layout: lanes 0-15 hold K={0,1}, lanes 16-31 hold K={2,3}
  const float* zp = z + (size_t)arow * CODE_DIM + 2 * half;
  v2f a_reg[64];
#pragma unroll
  for (int c = 0; c < 64; ++c) a_reg[c] = *(const v2f*)(zp + 4 * c);

  // --- cooperative codebook-tile copy assignment (16 floats / thread) ---
  const int cr = tid >> 4;          // code row within tile (0..15)
  const int cd = (tid & 15) * 16;   // d-chunk start (0..240)

#if VQ_ASYNC
  {
    const float* g = cb + (size_t)(0 * TILE_CODES + cr) * CODE_DIM + cd;
    float* l = &lds_b[0][cr * LDS_STRIDE + cd];
#pragma unroll
    for (int q = 0; q < 4; ++q)
      __builtin_amdgcn_global_load_async_to_lds_b128(
          (gas_v4i_ptr)(g + 4 * q), (las_v4i_ptr)(l + 4 * q), 0, 0);
  }
#endif

  float minv[8];
  int   mini[8];
#pragma unroll
  for (int r = 0; r < 8; ++r) { minv[r] = 3.4e38f; mini[r] = 0x7fffffff; }

  for (int kt = 0; kt < NUM_TILES; ++kt) {
    const int buf = kt & 1;
#if VQ_ASYNC
    if (kt + 1 < NUM_TILES) {       // prefetch next tile into other buffer
      const float* g = cb + (size_t)((kt + 1) * TILE_CODES + cr) * CODE_DIM + cd;
      float* l = &lds_b[buf ^ 1][cr * LDS_STRIDE + cd];
#pragma unroll
      for (int q = 0; q < 4; ++q)
        __builtin_amdgcn_global_load_async_to_lds_b128(
            (gas_v4i_ptr)(g + 4 * q), (las_v4i_ptr)(l + 4 * q), 0, 0);
      __builtin_amdgcn_s_wait_asynccnt(4);  // current tile's 4 loads done
    } else {
      __builtin_amdgcn_s_wait_asynccnt(0);
    }
    __syncthreads();
#else
    __syncthreads();
    {
      const float4* g = (const float4*)(cb + (size_t)(kt * TILE_CODES + cr) * CODE_DIM + cd);
      float4* l = (float4*)&lds_b[buf][cr * LDS_STRIDE + cd];
#pragma unroll
      for (int q = 0; q < 4; ++q) l[q] = g[q];
    }
    __syncthreads();
#endif

    const float en = enorm[kt * TILE_CODES + lane_lo];
    // B 4x16 f32 layout mirrors A: lane n<16 -> code n, K={0,1}; lane 16+n -> K={2,3}
    const float* bp = &lds_b[buf][lane_lo * LDS_STRIDE + 2 * half];

    v8f acc0 = {};
    v8f acc1 = {};
    // depth-2 software pipeline on the LDS B reads: keep two ds loads in
    // flight so the scheduler waits on the older load, not the newest.
    v2f b0 = *(const v2f*)(bp + 0);
    v2f b1 = *(const v2f*)(bp + 4);
#pragma unroll
    for (int c = 0; c < 64; c += 2) {
      v2f nb0 = b0, nb1 = b1;
      if (c + 2 < 64) {
        nb0 = *(const v2f*)(bp + 4 * (c + 2));
        nb1 = *(const v2f*)(bp + 4 * (c + 3));
      }
      acc0 = __builtin_amdgcn_wmma_f32_16x16x4_f32(
          false, a_reg[c],     false, b0, (short)0, acc0, false, false);
      acc1 = __builtin_amdgcn_wmma_f32_16x16x4_f32(
          false, a_reg[c + 1], false, b1, (short)0, acc1, false, false);
      b0 = nb0;
      b1 = nb1;
    }

    const int code = kt * TILE_CODES + lane_lo;
#pragma unroll
    for (int r = 0; r < 8; ++r) {
      float dist = en - 2.0f * (acc0[r] + acc1[r]);
      if (dist < minv[r]) { minv[r] = dist; mini[r] = code; }
    }

#if VQ_ASYNC
    __syncthreads();  // readers done before buffer is rewritten by next prefetch
#endif
  }

  // --- cross-lane argmin inside each 16-lane half (C layout: VGPR r ->
  //     M=r for lanes 0-15, M=r+8 for lanes 16-31) ---
#pragma unroll
  for (int r = 0; r < 8; ++r) {
#pragma unroll
    for (int off = 1; off <= 8; off <<= 1) {
      float ov = __shfl_xor(minv[r], off, 32);
      int   oi = __shfl_xor(mini[r], off, 32);
      if (ov < minv[r] || (ov == minv[r] && oi < mini[r])) {
        minv[r] = ov; mini[r] = oi;
      }
    }
  }
  if (lane_lo == 0) {  // lanes 0 and 16 hold the reduced results
    const int mrow = wrow + half * 8;
#pragma unroll
    for (int r = 0; r < 8; ++r) out_idx[mrow + r] = mini[r];
  }
}

// =====================================================================
// Kernel 3: gather z_q, write outputs, deterministic per-block loss partials
//   256 blocks x 256 threads; each iteration handles one full row (coalesced,
//   scalar index load).
// =====================================================================
__global__ __launch_bounds__(256) void vq_gather(const float* __restrict__ z,
                                                 const float* __restrict__ cb,
                                                 const int* __restrict__ idx,
                                                 float* __restrict__ out_zq,
                                                 float* __restrict__ out_idx_f,
                                                 float* __restrict__ partial) {
  __shared__ float red[256];
  const int t = threadIdx.x;
  const int rowbase = blockIdx.x * 128;
  float s = 0.f;
  for (int i = 0; i < 128; ++i) {
    const int row = rowbase + i;
    const int k = idx[row];
    const float c = cb[(size_t)k * CODE_DIM + t];
    const float zv = z[(size_t)row * CODE_DIM + t];
    out_zq[(size_t)row * CODE_DIM + t] = c;  // z_q_st == z_q numerically
    const float d = c - zv;
    s += d * d;
    if (t == 0) out_idx_f[row] = (float)k;
  }
  red[t] = s;
  __syncthreads();
  for (int off = 128; off >= 1; off >>= 1) {
    if (t < off) red[t] += red[t + off];
    __syncthreads();
  }
  if (t == 0) partial[blockIdx.x] = red[0];
}

// =====================================================================
// Kernel 4: fixed-order final loss reduction (replay-deterministic)
//   loss = codebook_loss + 0.25*commitment = 1.25 * mean((z_q - z)^2)
// =====================================================================
__global__ __launch_bounds__(256) void vq_finalize(const float* __restrict__ partial,
                                                   float* __restrict__ loss_out) {
  __shared__ float red[256];
  const int t = threadIdx.x;
  red[t] = partial[t];
  __syncthreads();
  for (int off = 128; off >= 1; off >>= 1) {
    if (t < off) red[t] += red[t + off];
    __syncthreads();
  }
  if (t == 0) *loss_out = 1.25f * red[0] / (float)N_ELEMS;
}

// =====================================================================
extern "C" void kernel_launch(void* const* d_in, const int* in_sizes, int n_in,
                              void* d_out, int out_size, void* d_ws, size_t ws_size,
                              hipStream_t stream) {
  const float* z  = (const float*)d_in[0];   // [32,1024,256] f32
  const float* cb = (const float*)d_in[1];   // [8192,256]    f32

  float* out       = (float*)d_out;
  float* out_zq    = out;                    // N_ELEMS floats
  float* loss      = out + N_ELEMS;          // 1 float
  float* out_idx_f = out + N_ELEMS + 1;      // N_ROWS floats

  float* ws      = (float*)d_ws;
  float* enorm   = ws;                        // 8192 floats
  int*   idx     = (int*)(ws + NUM_ENTRIES);  // 32768 ints
  float* partial = ws + NUM_ENTRIES + N_ROWS; // 256 floats

  vq_prep    <<<NUM_ENTRIES / 8, 256, 0, stream>>>(cb, enorm);
  vq_main    <<<N_ROWS / ROWS_PER_BLOCK, 256, 0, stream>>>(z, cb, enorm, idx);
  vq_gather  <<<256, 256, 0, stream>>>(z, cb, idx, out_zq, out_idx_f, partial);
  vq_finalize<<<1, 256, 0, stream>>>(partial, loss);
}